// TJSigNet_4896262717791
// MI455X (gfx1250) — compile-verified
//
#include <hip/hip_runtime.h>
#include <hip/hip_bf16.h>

typedef __attribute__((ext_vector_type(2))) float v2f;
typedef __attribute__((ext_vector_type(8))) float v8f;

#define NB    64      // batch
#define VM    34      // 17 * 2
#define NPATH (NB*VM) // 2176
#define T_LEN 300
#define NSEG  299
#define CCH   8       // time chunks per path (Chen composition is associative)
#define SIGC  120
#define K1    4080    // 34 * 120
#define N1    512
#define N2    155

// input (B, 3, 300, 17, 2) strides in floats
#define S_B 30600
#define S_C 10200
#define S_T 34

// ---------------------------------------------------------------------------
// Kernel 1: per-chunk depth-4 signature. Thread q = c*NPATH + p computes the
// signature of segments [299c/8, 299(c+1)/8) of path p, state in registers.
// chunks[p][c][120], 120 = [c1(3) | c2(9) | c3(27) | c4(81)]
// ---------------------------------------------------------------------------
__global__ __launch_bounds__(128)
void tjsig_chunk_kernel(const float* __restrict__ inp, float* __restrict__ chunks)
{
    const int q  = blockIdx.x * 128 + threadIdx.x;   // exactly NPATH*CCH threads
    const int p  = q % NPATH;                        // lanes: consecutive paths -> coalesced
    const int cc = q / NPATH;                        // chunk id (uniform per wave)
    const int b  = p / VM;
    const int vm = p - b * VM;
    const float* __restrict__ base = inp + (size_t)b * S_B + vm;

    const int s = (NSEG * cc) / CCH;                 // first segment of chunk
    const int e = (NSEG * (cc + 1)) / CCH;           // one past last segment

    float c1[3], c2[9], c3[27], c4[81];
#pragma unroll
    for (int i = 0; i < 3; ++i)  c1[i] = 0.f;
#pragma unroll
    for (int i = 0; i < 9; ++i)  c2[i] = 0.f;
#pragma unroll
    for (int i = 0; i < 27; ++i) c3[i] = 0.f;
#pragma unroll
    for (int i = 0; i < 81; ++i) c4[i] = 0.f;

    float prev0 = base[s * S_T];
    float prev1 = base[S_C + s * S_T];
    float prev2 = base[2 * S_C + s * S_T];

    for (int t = s + 1; t <= e; ++t) {
        const float x0 = base[t * S_T];
        const float x1 = base[S_C + t * S_T];
        const float x2 = base[2 * S_C + t * S_T];
        float dx[3];
        dx[0] = x0 - prev0; dx[1] = x1 - prev1; dx[2] = x2 - prev2;
        prev0 = x0; prev1 = x1; prev2 = x2;

        // per-segment exp(dx) levels: g2 = dx^2/2!, g3 = dx^3/3! (g4 folded below)
        float g2[9], g3[27];
#pragma unroll
        for (int i = 0; i < 3; ++i)
#pragma unroll
            for (int j = 0; j < 3; ++j)
                g2[i * 3 + j] = dx[i] * dx[j] * 0.5f;
#pragma unroll
        for (int ij = 0; ij < 9; ++ij)
#pragma unroll
            for (int k = 0; k < 3; ++k)
                g3[ij * 3 + k] = g2[ij] * dx[k] * (1.f / 3.f);

        // Chen: update highest level first (uses old lower levels)
        // c4 += c3 (x) dx  +  g3 (x) dx / 4   (g4 = g3 (x) dx / 4)
#pragma unroll
        for (int ijk = 0; ijk < 27; ++ijk) {
            const float a = c3[ijk];
            const float g = g3[ijk] * 0.25f;
#pragma unroll
            for (int l = 0; l < 3; ++l)
                c4[ijk * 3 + l] += (a + g) * dx[l];
        }
#pragma unroll
        for (int ij = 0; ij < 9; ++ij)
#pragma unroll
            for (int kl = 0; kl < 9; ++kl)
                c4[ij * 9 + kl] += c2[ij] * g2[kl];
#pragma unroll
        for (int i = 0; i < 3; ++i)
#pragma unroll
            for (int jkl = 0; jkl < 27; ++jkl)
                c4[i * 27 + jkl] += c1[i] * g3[jkl];

        // c3 += c2 (x) dx + c1 (x) g2 + g3
#pragma unroll
        for (int ij = 0; ij < 9; ++ij)
#pragma unroll
            for (int k = 0; k < 3; ++k)
                c3[ij * 3 + k] += c2[ij] * dx[k];
#pragma unroll
        for (int i = 0; i < 3; ++i)
#pragma unroll
            for (int jk = 0; jk < 9; ++jk)
                c3[i * 9 + jk] += c1[i] * g2[jk];
#pragma unroll
        for (int x = 0; x < 27; ++x)
            c3[x] += g3[x];

        // c2 += c1 (x) dx + g2
#pragma unroll
        for (int i = 0; i < 3; ++i)
#pragma unroll
            for (int j = 0; j < 3; ++j)
                c2[i * 3 + j] += c1[i] * dx[j] + g2[i * 3 + j];

        // c1 += dx
#pragma unroll
        for (int i = 0; i < 3; ++i)
            c1[i] += dx[i];
    }

    float* __restrict__ o = chunks + ((size_t)p * CCH + cc) * SIGC;
#pragma unroll
    for (int i = 0; i < 3; ++i)  o[i]      = c1[i];
#pragma unroll
    for (int i = 0; i < 9; ++i)  o[3 + i]  = c2[i];
#pragma unroll
    for (int i = 0; i < 27; ++i) o[12 + i] = c3[i];
#pragma unroll
    for (int i = 0; i < 81; ++i) o[39 + i] = c4[i];
}

// ---------------------------------------------------------------------------
// Kernel 2: fold the CCH chunk signatures of each path with tensor-level
// Chen composition: n_k = c_k + sum_{i+j=k} c_i (x) g_j. One thread per path.
// Level-4 of the incoming chunk is streamed from memory to cap VGPR use.
// ---------------------------------------------------------------------------
__global__ __launch_bounds__(128)
void tjsig_combine_kernel(const float* __restrict__ chunks, float* __restrict__ sig)
{
    const int p  = blockIdx.x * 128 + threadIdx.x;   // exactly NPATH threads
    const int b  = p / VM;
    const int vm = p - b * VM;
    const float* __restrict__ ch0 = chunks + (size_t)p * CCH * SIGC;

    float c1[3], c2[9], c3[27], c4[81];
#pragma unroll
    for (int i = 0; i < 3; ++i)  c1[i] = ch0[i];
#pragma unroll
    for (int i = 0; i < 9; ++i)  c2[i] = ch0[3 + i];
#pragma unroll
    for (int i = 0; i < 27; ++i) c3[i] = ch0[12 + i];
#pragma unroll
    for (int i = 0; i < 81; ++i) c4[i] = ch0[39 + i];

    for (int cc = 1; cc < CCH; ++cc) {
        const float* __restrict__ g = ch0 + (size_t)cc * SIGC;
        float g1[3], g2[9], g3[27];
#pragma unroll
        for (int i = 0; i < 3; ++i)  g1[i] = g[i];
#pragma unroll
        for (int i = 0; i < 9; ++i)  g2[i] = g[3 + i];
#pragma unroll
        for (int i = 0; i < 27; ++i) g3[i] = g[12 + i];

        // c4 = c4 + c3(x)g1 + c2(x)g2 + c1(x)g3 + g4  (g4 streamed from memory)
#pragma unroll
        for (int ijk = 0; ijk < 27; ++ijk)
#pragma unroll
            for (int l = 0; l < 3; ++l)
                c4[ijk * 3 + l] += c3[ijk] * g1[l];
#pragma unroll
        for (int ij = 0; ij < 9; ++ij)
#pragma unroll
            for (int kl = 0; kl < 9; ++kl)
                c4[ij * 9 + kl] += c2[ij] * g2[kl];
#pragma unroll
        for (int i = 0; i < 3; ++i)
#pragma unroll
            for (int jkl = 0; jkl < 27; ++jkl)
                c4[i * 27 + jkl] += c1[i] * g3[jkl];
#pragma unroll
        for (int x = 0; x < 81; ++x)
            c4[x] += g[39 + x];

        // c3 = c3 + c2(x)g1 + c1(x)g2 + g3
#pragma unroll
        for (int ij = 0; ij < 9; ++ij)
#pragma unroll
            for (int k = 0; k < 3; ++k)
                c3[ij * 3 + k] += c2[ij] * g1[k];
#pragma unroll
        for (int i = 0; i < 3; ++i)
#pragma unroll
            for (int jk = 0; jk < 9; ++jk)
                c3[i * 9 + jk] += c1[i] * g2[jk];
#pragma unroll
        for (int x = 0; x < 27; ++x)
            c3[x] += g3[x];

        // c2 = c2 + c1(x)g1 + g2
#pragma unroll
        for (int i = 0; i < 3; ++i)
#pragma unroll
            for (int j = 0; j < 3; ++j)
                c2[i * 3 + j] += c1[i] * g1[j] + g2[i * 3 + j];

        // c1 = c1 + g1
#pragma unroll
        for (int i = 0; i < 3; ++i)
            c1[i] += g1[i];
    }

    float* __restrict__ o = sig + (size_t)b * K1 + vm * SIGC;
#pragma unroll
    for (int i = 0; i < 3; ++i)  o[i]      = c1[i];
#pragma unroll
    for (int i = 0; i < 9; ++i)  o[3 + i]  = c2[i];
#pragma unroll
    for (int i = 0; i < 27; ++i) o[12 + i] = c3[i];
#pragma unroll
    for (int i = 0; i < 81; ++i) o[39 + i] = c4[i];
}

// ---------------------------------------------------------------------------
// Kernels 3/4: out[M x Nout] = A[M x K] * W[Nout x K]^T + bias, via
// V_WMMA_F32_16X16X4_F32 (full f32 precision — workload is memory bound,
// no reason to drop precision). One 16x16 output tile per wave.
//
// A frag (16x4 f32): lane l<16 -> M=l, K={0,1}; lanes 16-31 -> K={2,3}
// B frag (4x16 f32): VGPR j -> K = j + (lane/16)*2, N = lane%16
// C/D   (16x16 f32): VGPR r -> M = r + (lane/16)*8, N = lane%16
// ---------------------------------------------------------------------------
__global__ __launch_bounds__(128)
void tjsig_gemm_wmma_bias(const float* __restrict__ A, int K,
                          const float* __restrict__ W,       // (Nout x K) row-major
                          const float* __restrict__ bias,    // (Nout)
                          float* __restrict__ out, int Nout,
                          int mtiles)
{
    const int tid  = blockIdx.x * 128 + threadIdx.x;
    const int wave = tid >> 5;       // wave32
    const int lane = tid & 31;
    const int mt   = wave % mtiles;
    const int nt   = wave / mtiles;
    const int half = lane >> 4;      // which half-wave
    const int l16  = lane & 15;

    const int m = mt * 16 + l16;                 // A row for this lane
    const int n = nt * 16 + l16;                 // output column for this lane
    const int nw = (n < Nout) ? n : (Nout - 1);  // branch-free OOB clamp (EXEC stays all-1)

    const float* __restrict__ arow = A + (size_t)m * K;
    const float* __restrict__ wrow = W + (size_t)nw * K;

    v8f acc = {0.f, 0.f, 0.f, 0.f, 0.f, 0.f, 0.f, 0.f};

#pragma unroll 4
    for (int k0 = 0; k0 < K; k0 += 4) {
        const int ka = k0 + (half << 1);
        v2f a, bb;
        a.x  = arow[ka];
        a.y  = arow[ka + 1];
        bb.x = wrow[ka];
        bb.y = wrow[ka + 1];
        acc = __builtin_amdgcn_wmma_f32_16x16x4_f32(
            /*neg_a=*/false, a, /*neg_b=*/false, bb,
            /*c_mod=*/(short)0, acc, /*reuse_a=*/false, /*reuse_b=*/false);
    }

    const float bv = bias[nw];
#pragma unroll
    for (int r = 0; r < 8; ++r) {
        const int mrow = mt * 16 + (half << 3) + r;
        if (n < Nout)
            out[(size_t)mrow * Nout + n] = acc[r] + bv;
    }
}

// ---------------------------------------------------------------------------
extern "C" void kernel_launch(void* const* d_in, const int* in_sizes, int n_in,
                              void* d_out, int out_size, void* d_ws, size_t ws_size,
                              hipStream_t stream) {
    const float* inp = (const float*)d_in[0];   // (64, 3, 300, 17, 2)
    const float* W1  = (const float*)d_in[1];   // (512, 4080)
    const float* b1  = (const float*)d_in[2];   // (512)
    const float* W2  = (const float*)d_in[3];   // (155, 512)
    const float* b2  = (const float*)d_in[4];   // (155)
    float* out = (float*)d_out;                 // (64, 155)

    float* chunks = (float*)d_ws;                       // (2176, 8, 120)
    float* sig    = chunks + (size_t)NPATH * CCH * SIGC; // (64, 4080)
    float* h      = sig + (size_t)NB * K1;               // (64, 512)

    // 1) chunked signatures: 2176*8 threads = 136 blocks x 128 (exact fit)
    tjsig_chunk_kernel<<<(NPATH * CCH) / 128, 128, 0, stream>>>(inp, chunks);

    // 2) Chen-fold 8 chunks per path: 2176 threads = 17 blocks x 128 (exact fit)
    tjsig_combine_kernel<<<NPATH / 128, 128, 0, stream>>>(chunks, sig);

    // 3) h = sig @ W1^T + b1 : mtiles=4, ntiles=32 -> 128 waves -> 32 blocks
    tjsig_gemm_wmma_bias<<<(4 * (N1 / 16) * 32) / 128, 128, 0, stream>>>(
        sig, K1, W1, b1, h, N1, 4);

    // 4) out = h @ W2^T + b2 : mtiles=4, ntiles=10 (155 cols) -> 40 waves -> 10 blocks
    tjsig_gemm_wmma_bias<<<(4 * ((N2 + 15) / 16) * 32) / 128, 128, 0, stream>>>(
        h, N1, W2, b2, out, N2, 4);
}